// TransformerGNNLayer_65008624992408
// MI455X (gfx1250) — compile-verified
//
#include <hip/hip_runtime.h>
#include <math.h>

typedef float v2f __attribute__((ext_vector_type(2)));
typedef float v8f __attribute__((ext_vector_type(8)));

#define HD    128   // heads * dim
#define NHEAD 8
#define DHEAD 16
#define INDIM 128
#define EDIM  32

// ---------------------------------------------------------------------------
// f32 WMMA GEMM with register-resident B fragments:
//   C[M,NCOLS] = A[M,K] @ B[K,NCOLS] (+ bias[NCOLS])
// Each wave owns one 16-wide column strip (tn): it preloads ALL B fragments
// for that strip into registers (K/4 x v2f; 64 VGPRs for K=128), then
// iterates over 16-row M tiles.  Inner loop = 1 global_load_b64 (A frag,
// contiguous v2f) + 1 V_WMMA_F32_16X16X4_F32 per K-step.
// Fragment layouts per ISA 7.12.2:
//   A 16x4: lanes 0-15 hold K=kk,kk+1; lanes 16-31 hold K=kk+2,kk+3
//   B 4x16: same split across the K rows, N striped across lanes
//   C/D:    8 VGPRs, M = r + 8*(lane>=16), N = lane&15
// ---------------------------------------------------------------------------
template<int K, int NCOLS>
__global__ void wmma_gemm_bias(const float* __restrict__ A,
                               const float* __restrict__ B,
                               const float* __restrict__ bias,
                               float* __restrict__ C,
                               int M) {
    constexpr int KSTEPS = K / 4;
    constexpr int NTILES = NCOLS / 16;

    const int lane = threadIdx.x & 31;
    const int hl   = lane >> 4;          // 0: lanes 0-15, 1: lanes 16-31
    const int l    = lane & 15;
    const int waveId = (blockIdx.x * blockDim.x + threadIdx.x) >> 5;
    const int nWaves = (gridDim.x * blockDim.x) >> 5;
    const int wavesPerCol = nWaves / NTILES;          // launch: nWaves % NTILES == 0
    const int tn  = waveId / wavesPerCol;
    const int tm0 = waveId % wavesPerCol;
    if (tn >= NTILES) return;

    const int col    = (tn << 4) + l;
    const int mTiles = (M + 15) >> 4;

    // ---- B fragments for this column strip: resident in registers ----
    v2f bfrag[KSTEPS];
#pragma unroll
    for (int s = 0; s < KSTEPS; ++s) {
        const int ka = (s << 2) + (hl << 1);
        v2f b;
        b.x = B[(size_t)ka * NCOLS + col];
        b.y = B[(size_t)(ka + 1) * NCOLS + col];
        bfrag[s] = b;
    }
    const float bz = bias ? bias[col] : 0.f;

    for (int tm = tm0; tm < mTiles; tm += wavesPerCol) {
        int rowA = (tm << 4) + l;
        if (rowA >= M) rowA = M - 1;       // clamped load; store guarded
        const float* arow = A + (size_t)rowA * K + (hl << 1);

        v8f acc = {0.f, 0.f, 0.f, 0.f, 0.f, 0.f, 0.f, 0.f};
#pragma unroll
        for (int s = 0; s < KSTEPS; ++s) {
            const v2f a = *(const v2f*)(arow + (s << 2));   // 8B-aligned b64 load
            acc = __builtin_amdgcn_wmma_f32_16x16x4_f32(
                false, a, false, bfrag[s], (short)0, acc, false, false);
        }

#pragma unroll
        for (int r = 0; r < 8; ++r) {
            const int row = (tm << 4) + r + (hl << 3);
            if (row < M)
                C[(size_t)row * NCOLS + col] = acc[r] + bz;
        }
    }
}

// ---------------------------------------------------------------------------
// Workspace init: agg = 0, per-(node,head) max = -inf, denom = 0
// ---------------------------------------------------------------------------
__global__ void init_buffers(float* __restrict__ agg,
                             float* __restrict__ mx,
                             float* __restrict__ den,
                             int n_agg, int n_nh) {
    int i = blockIdx.x * blockDim.x + threadIdx.x;
    const int stride = gridDim.x * blockDim.x;
    for (; i < n_agg; i += stride) {
        agg[i] = 0.f;
        if (i < n_nh) {
            mx[i]  = -3.402823466e+38f;
            den[i] = 0.f;
        }
    }
}

__device__ __forceinline__ void atomicMaxFloat(float* addr, float val) {
    int* iaddr = (int*)addr;
    int old = __float_as_int(*addr);
    while (__int_as_float(old) < val) {
        int assumed = old;
        old = atomicCAS(iaddr, assumed, __float_as_int(val));
        if (old == assumed) break;
    }
}

// ---------------------------------------------------------------------------
// Per-(edge,head) attention logit + running segment max (atomic)
// alpha[e,h] = <q[dst,h,:], k[src,h,:] + eproj[e,h,:]> / sqrt(16)
// ---------------------------------------------------------------------------
__global__ void edge_logits(const float* __restrict__ q,
                            const float* __restrict__ k,
                            const float* __restrict__ eproj,
                            const long long* __restrict__ eidx,
                            float* __restrict__ logits,
                            float* __restrict__ mx,
                            int E) {
    int i = blockIdx.x * blockDim.x + threadIdx.x;
    const int stride = gridDim.x * blockDim.x;
    const int total = E * NHEAD;
    for (; i < total; i += stride) {
        const int e = i >> 3;
        const int h = i & 7;
        const long long src = eidx[e];
        const long long dst = eidx[(size_t)E + e];
        const float4* qp = (const float4*)(q + (size_t)dst * HD + h * DHEAD);
        const float4* kp = (const float4*)(k + (size_t)src * HD + h * DHEAD);
        const float4* ep = (const float4*)(eproj + (size_t)e * HD + h * DHEAD);
        float s = 0.f;
#pragma unroll
        for (int j = 0; j < 4; ++j) {
            const float4 qv = qp[j], kv = kp[j], ev = ep[j];
            s += qv.x * (kv.x + ev.x) + qv.y * (kv.y + ev.y) +
                 qv.z * (kv.z + ev.z) + qv.w * (kv.w + ev.w);
        }
        s *= 0.25f;                       // 1/sqrt(D), D=16
        logits[i] = s;
        atomicMaxFloat(&mx[(size_t)dst * NHEAD + h], s);
    }
}

// ---------------------------------------------------------------------------
// ex = exp(logit - max[dst]); denom[dst] += ex (atomic); logits <- ex
// ---------------------------------------------------------------------------
__global__ void edge_exp(const long long* __restrict__ eidx,
                         const float* __restrict__ mx,
                         float* __restrict__ logits,
                         float* __restrict__ den,
                         int E) {
    int i = blockIdx.x * blockDim.x + threadIdx.x;
    const int stride = gridDim.x * blockDim.x;
    const int total = E * NHEAD;
    for (; i < total; i += stride) {
        const int e = i >> 3;
        const int h = i & 7;
        const long long dst = eidx[(size_t)E + e];
        const float ex = __expf(logits[i] - mx[(size_t)dst * NHEAD + h]);
        logits[i] = ex;
        atomicAdd(&den[(size_t)dst * NHEAD + h], ex);
    }
}

// ---------------------------------------------------------------------------
// agg[dst,h,d] += (ex / denom) * (v[src,h,d] + eproj[e,h,d])
// One thread per (edge, group-of-4 channels): 32 threads = one edge row.
// ---------------------------------------------------------------------------
__global__ void edge_aggregate(const float* __restrict__ v,
                               const float* __restrict__ eproj,
                               const long long* __restrict__ eidx,
                               const float* __restrict__ ex,
                               const float* __restrict__ den,
                               float* __restrict__ agg,
                               int E) {
    int i = blockIdx.x * blockDim.x + threadIdx.x;
    const int stride = gridDim.x * blockDim.x;
    const int total = E * 32;
    for (; i < total; i += stride) {
        const int e  = i >> 5;
        const int hd = (i & 31) << 2;     // 0..124
        const int h  = hd >> 4;
        const long long src = eidx[e];
        const long long dst = eidx[(size_t)E + e];
        const float coef = ex[(size_t)e * NHEAD + h] /
                           (den[(size_t)dst * NHEAD + h] + 1e-16f);
        const float4 vv = *(const float4*)(v + (size_t)src * HD + hd);
        const float4 ee = *(const float4*)(eproj + (size_t)e * HD + hd);
        float* ap = agg + (size_t)dst * HD + hd;
        atomicAdd(ap + 0, coef * (vv.x + ee.x));
        atomicAdd(ap + 1, coef * (vv.y + ee.y));
        atomicAdd(ap + 2, coef * (vv.z + ee.z));
        atomicAdd(ap + 3, coef * (vv.w + ee.w));
    }
}

// ---------------------------------------------------------------------------
// out = LayerNorm(leaky_relu(agg + skip)) * gamma + beta
// One wave32 per node, 4 channels per lane, shfl_xor butterfly reduction.
// ---------------------------------------------------------------------------
__global__ void finalize(const float* __restrict__ agg,
                         const float* __restrict__ skip,
                         const float* __restrict__ gamma,
                         const float* __restrict__ beta,
                         float* __restrict__ out,
                         int N) {
    const int lane = threadIdx.x & 31;
    const int wid  = (blockIdx.x * blockDim.x + threadIdx.x) >> 5;
    const int nw   = (gridDim.x * blockDim.x) >> 5;
    const int hd   = lane << 2;

    for (int n = wid; n < N; n += nw) {
        const float4 a = *(const float4*)(agg + (size_t)n * HD + hd);
        const float4 s = *(const float4*)(skip + (size_t)n * HD + hd);
        float v0 = a.x + s.x, v1 = a.y + s.y, v2 = a.z + s.z, v3 = a.w + s.w;
        v0 = v0 > 0.f ? v0 : 0.2f * v0;
        v1 = v1 > 0.f ? v1 : 0.2f * v1;
        v2 = v2 > 0.f ? v2 : 0.2f * v2;
        v3 = v3 > 0.f ? v3 : 0.2f * v3;

        float sum = v0 + v1 + v2 + v3;
        float sq  = v0 * v0 + v1 * v1 + v2 * v2 + v3 * v3;
#pragma unroll
        for (int m = 16; m > 0; m >>= 1) {
            sum += __shfl_xor(sum, m, 32);
            sq  += __shfl_xor(sq, m, 32);
        }
        const float mu  = sum * (1.f / HD);
        const float var = sq * (1.f / HD) - mu * mu;
        const float rs  = rsqrtf(var + 1e-5f);

        const float4 g = *(const float4*)(gamma + hd);
        const float4 b = *(const float4*)(beta + hd);
        float4 o;
        o.x = (v0 - mu) * rs * g.x + b.x;
        o.y = (v1 - mu) * rs * g.y + b.y;
        o.z = (v2 - mu) * rs * g.z + b.z;
        o.w = (v3 - mu) * rs * g.w + b.w;
        *(float4*)(out + (size_t)n * HD + hd) = o;
    }
}

// ---------------------------------------------------------------------------
extern "C" void kernel_launch(void* const* d_in, const int* in_sizes, int n_in,
                              void* d_out, int out_size, void* d_ws, size_t ws_size,
                              hipStream_t stream) {
    const float*     x     = (const float*)d_in[0];
    const long long* eidx  = (const long long*)d_in[1];
    const float*     eattr = (const float*)d_in[2];
    const float*     Wq    = (const float*)d_in[3];
    const float*     bq    = (const float*)d_in[4];
    const float*     Wk    = (const float*)d_in[5];
    const float*     bk    = (const float*)d_in[6];
    const float*     Wv    = (const float*)d_in[7];
    const float*     bv    = (const float*)d_in[8];
    const float*     We    = (const float*)d_in[9];
    const float*     Wskip = (const float*)d_in[10];
    const float*     bskip = (const float*)d_in[11];
    const float*     gamma = (const float*)d_in[12];
    const float*     beta  = (const float*)d_in[13];
    float*           out   = (float*)d_out;

    const int N = in_sizes[0] / INDIM;
    const int E = in_sizes[2] / EDIM;

    // Workspace carve-out (floats)
    float* ws = (float*)d_ws;
    size_t off = 0;
    float* q      = ws + off; off += (size_t)N * HD;
    float* k      = ws + off; off += (size_t)N * HD;
    float* v      = ws + off; off += (size_t)N * HD;
    float* skip   = ws + off; off += (size_t)N * HD;
    float* agg    = ws + off; off += (size_t)N * HD;
    float* mx     = ws + off; off += (size_t)N * NHEAD;
    float* den    = ws + off; off += (size_t)N * NHEAD;
    float* logits = ws + off; off += (size_t)E * NHEAD;
    float* eproj  = ws + off; off += (size_t)E * HD;

    // ---- Dense projections via WMMA ------------------------------------
    // 256 threads = 8 waves per block; nWaves multiple of NTILES(=8) always.
    {
        const int blocks = 512;                       // 4096 waves, 512 per column strip
        wmma_gemm_bias<INDIM, HD><<<blocks, 256, 0, stream>>>(x, Wq,    bq,    q,    N);
        wmma_gemm_bias<INDIM, HD><<<blocks, 256, 0, stream>>>(x, Wk,    bk,    k,    N);
        wmma_gemm_bias<INDIM, HD><<<blocks, 256, 0, stream>>>(x, Wv,    bv,    v,    N);
        wmma_gemm_bias<INDIM, HD><<<blocks, 256, 0, stream>>>(x, Wskip, bskip, skip, N);
    }
    {
        const int blocks = 2048;                      // 16384 waves, 2048 per column strip
        wmma_gemm_bias<EDIM, HD><<<blocks, 256, 0, stream>>>(eattr, We, nullptr, eproj, E);
    }

    // ---- Segment softmax + aggregation ---------------------------------
    init_buffers<<<(N * HD + 255) / 256, 256, 0, stream>>>(agg, mx, den, N * HD, N * NHEAD);
    edge_logits<<<(E * NHEAD + 255) / 256, 256, 0, stream>>>(q, k, eproj, eidx, logits, mx, E);
    edge_exp<<<(E * NHEAD + 255) / 256, 256, 0, stream>>>(eidx, mx, logits, den, E);
    edge_aggregate<<<(E * 32 + 255) / 256, 256, 0, stream>>>(v, eproj, eidx, logits, den, agg, E);

    // ---- Skip + leaky ReLU + LayerNorm ---------------------------------
    finalize<<<(N * 32 + 255) / 256, 256, 0, stream>>>(agg, skip, gamma, beta, out, N);
}